// RopeEmbeddings_27616639713679
// MI455X (gfx1250) — compile-verified
//
#include <hip/hip_runtime.h>
#include <math.h>

// RoPE (reference uses theta = 10000^(+2i/d)) over X[N=32, D=128, L=8192] f32.
// Pure streaming op: 256 MB total traffic -> ~11 us floor @ 23.3 TB/s.
// Strategy: thread = one (i, l..l+3) column chunk; sincos computed once,
// reused across all N=32 batches; b128 NT loads/stores; gfx1250
// global_prefetch_b8 for the 4MB-strided next-batch lines.

#define RP_N    32
#define RP_D    128
#define RP_HALF 64
#define RP_L    8192

typedef __attribute__((ext_vector_type(4))) float v4f;

__global__ __launch_bounds__(256) void rope_f32_kernel(const float* __restrict__ X,
                                                       float* __restrict__ Out)
{
    const int i  = blockIdx.y;                          // 0..63 (rotary pair index)
    const int t  = blockIdx.x * 256 + threadIdx.x;      // float4 index along L: 0..2047
    const int l0 = t << 2;                              // first l of this thread's 4

    // theta = 10000^(2i/128) = exp(i * ln(10000)/64)
    const float theta = __expf((float)i * 0.14391156536914104f);

    // One sincos per (i,l): computed once, reused for all 32 batches.
    float s0, s1, s2, s3, c0, c1, c2, c3;
    sincosf(theta * (float)(l0 + 0), &s0, &c0);
    sincosf(theta * (float)(l0 + 1), &s1, &c1);
    sincosf(theta * (float)(l0 + 2), &s2, &c2);
    sincosf(theta * (float)(l0 + 3), &s3, &c3);
    v4f cv = (v4f){c0, c1, c2, c3};
    v4f sv = (v4f){s0, s1, s2, s3};

    const v4f* __restrict__ Xv = reinterpret_cast<const v4f*>(X);
    v4f* __restrict__       Ov = reinterpret_cast<v4f*>(Out);

    const size_t nStride4 = (size_t)(RP_D * RP_L) / 4;      // 262144 float4 per batch
    size_t i1 = ((size_t)i * RP_L + (size_t)l0) >> 2;       // x1: row i
    size_t i2 = i1 + (size_t)(RP_HALF * RP_L) / 4;          // x2: row i+64

#pragma unroll 4
    for (int n = 0; n < RP_N; ++n) {
        // Prefetch next batch's lines (4 MB away: guaranteed cache miss).
        // Speculative global_prefetch_b8; OOB prefetches are silently dropped.
        __builtin_prefetch(&Xv[i1 + nStride4], 0, 0);
        __builtin_prefetch(&Xv[i2 + nStride4], 0, 0);

        // Streamed once, never reused -> non-temporal (TH=NT) b128 accesses.
        v4f a = __builtin_nontemporal_load(&Xv[i1]);
        v4f b = __builtin_nontemporal_load(&Xv[i2]);

        v4f o1, o2;
        o1 = cv * a - sv * b;     // out[:, i,    :] = c*x1 - s*x2
        o2 = sv * a + cv * b;     // out[:, i+64, :] = s*x1 + c*x2

        __builtin_nontemporal_store(o1, &Ov[i1]);
        __builtin_nontemporal_store(o2, &Ov[i2]);

        i1 += nStride4;
        i2 += nStride4;
    }
}

extern "C" void kernel_launch(void* const* d_in, const int* in_sizes, int n_in,
                              void* d_out, int out_size, void* d_ws, size_t ws_size,
                              hipStream_t stream)
{
    (void)in_sizes; (void)n_in; (void)out_size; (void)d_ws; (void)ws_size;

    const float* X = (const float*)d_in[0];
    float*     Out = (float*)d_out;

    // grid.x: L/(256 threads * 4 floats) = 8 tiles along L
    // grid.y: 64 rotary rows
    dim3 grid(RP_L / (256 * 4), RP_HALF, 1);
    dim3 block(256, 1, 1);
    rope_f32_kernel<<<grid, block, 0, stream>>>(X, Out);
}